// Neg_Pearson_17566416241182
// MI455X (gfx1250) — compile-verified
//
#include <hip/hip_runtime.h>

// ---------------------------------------------------------------------------
// Neg-Pearson over rows: 5 fused row reductions, memory-bound HBM stream
// (205 MB read-once -> ~8.8 us roofline at 23.3 TB/s).
// CDNA5 path: GLOBAL_LOAD_ASYNC_TO_LDS_B128 (ASYNCcnt) 4-stage pipeline +
// s_wait_asynccnt, NT load hints, wave32 shuffle reductions.
// ---------------------------------------------------------------------------

#define TPB    256   // 8 wave32s per block
#define BPR    8     // blocks per row -> 2048 blocks total
#define WAVE   32
#define STAGES 4     // async pipeline depth (8 b128 ops in flight per wave)

// LDS byte-offset of a pointer that statically points into __shared__ memory.
__device__ __forceinline__ unsigned lds_off_u32(const void* p) {
  return (unsigned)(unsigned long long)(const __attribute__((address_space(3))) void*)p;
}

// Async copy of 16 bytes/lane global -> LDS. Tracked by ASYNCcnt, not LOADcnt.
__device__ __forceinline__ void async_copy_b128(unsigned lds_off, const void* gaddr) {
  asm volatile("global_load_async_to_lds_b128 %0, %1, off th:TH_LOAD_NT"
               :: "v"(lds_off), "v"(gaddr)
               : "memory");
}

#define WAIT_ASYNCCNT(n) asm volatile("s_wait_asynccnt %0" :: "i"(n) : "memory")

__global__ void __launch_bounds__(TPB)
pearson_partial_kernel(const float* __restrict__ preds,
                       const float* __restrict__ labels,
                       float* __restrict__ partials,   // [B][BPR][5]
                       int n) {
  __shared__ float4 sx[STAGES][TPB];
  __shared__ float4 sy[STAGES][TPB];
  __shared__ float  wred[TPB / WAVE][5];

  const int tid = threadIdx.x;
  const int blk = blockIdx.x;       // 0..BPR-1
  const int row = blockIdx.y;       // 0..B-1
  const int n4  = n >> 2;           // float4 count per row
  const long long rowbase = (long long)row * n;
  const float4* p4 = (const float4*)(preds  + rowbase);
  const float4* l4 = (const float4*)(labels + rowbase);

  const int slots = BPR * TPB;                 // float4s consumed per chunk step
  const int nc    = (n4 + slots - 1) / slots;  // chunks this block iterates

  // Per-lane LDS staging slots (no cross-wave sharing -> no barriers needed;
  // each lane reads back only its own slot, each wave waits its own ASYNCcnt).
  unsigned lx[STAGES], ly[STAGES];
#pragma unroll
  for (int s = 0; s < STAGES; ++s) {
    lx[s] = lds_off_u32(&sx[s][tid]);
    ly[s] = lds_off_u32(&sy[s][tid]);
  }

  float s_x = 0.f, s_y = 0.f, s_xy = 0.f, s_xx = 0.f, s_yy = 0.f;

  auto issue = [&](int c, int st) {
    int idx4 = (c * BPR + blk) * TPB + tid;
    int i4c  = idx4 < (n4 - 1) ? idx4 : (n4 - 1);   // clamp: every lane issues,
    async_copy_b128(lx[st], p4 + i4c);              // keeps ASYNCcnt exact
    async_copy_b128(ly[st], l4 + i4c);
  };

  // Prologue: fill the ring (up to STAGES-1 chunks ahead + current).
#pragma unroll
  for (int s = 0; s < STAGES - 1; ++s)
    if (s < nc) issue(s, s);

  for (int c = 0; c < nc; ++c) {
    const int cur = c & (STAGES - 1);
    if (c + STAGES - 1 < nc) {
      issue(c + STAGES - 1, (c + STAGES - 1) & (STAGES - 1));
      WAIT_ASYNCCNT(2 * (STAGES - 1));  // chunk c's pair retired, rest in flight
    } else {
      WAIT_ASYNCCNT(0);                 // drain phase (last STAGES-1 iters only)
    }
    const float4 x = sx[cur][tid];
    const float4 y = sy[cur][tid];
    const int idx4 = (c * BPR + blk) * TPB + tid;
    if (idx4 < n4) {
      s_x += (x.x + x.y) + (x.z + x.w);
      s_y += (y.x + y.y) + (y.z + y.w);
      s_xy = fmaf(x.x, y.x, s_xy); s_xy = fmaf(x.y, y.y, s_xy);
      s_xy = fmaf(x.z, y.z, s_xy); s_xy = fmaf(x.w, y.w, s_xy);
      s_xx = fmaf(x.x, x.x, s_xx); s_xx = fmaf(x.y, x.y, s_xx);
      s_xx = fmaf(x.z, x.z, s_xx); s_xx = fmaf(x.w, x.w, s_xx);
      s_yy = fmaf(y.x, y.x, s_yy); s_yy = fmaf(y.y, y.y, s_yy);
      s_yy = fmaf(y.z, y.z, s_yy); s_yy = fmaf(y.w, y.w, s_yy);
    }
  }

  // Scalar tail (n % 4 elements), handled once per row by block 0.
  if (blk == 0) {
    for (int e = (n4 << 2) + tid; e < n; e += TPB) {
      const float x = preds[rowbase + e];
      const float y = labels[rowbase + e];
      s_x += x; s_y += y;
      s_xy = fmaf(x, y, s_xy);
      s_xx = fmaf(x, x, s_xx);
      s_yy = fmaf(y, y, s_yy);
    }
  }

  // Wave32 shuffle reduction, then cross-wave via LDS.
  float v[5] = { s_x, s_y, s_xy, s_xx, s_yy };
#pragma unroll
  for (int k = 0; k < 5; ++k)
#pragma unroll
    for (int off = WAVE / 2; off > 0; off >>= 1)
      v[k] += __shfl_down(v[k], off, WAVE);

  const int wave = tid / WAVE, lane = tid % WAVE;
  if (lane == 0) {
#pragma unroll
    for (int k = 0; k < 5; ++k) wred[wave][k] = v[k];
  }
  __syncthreads();
  if (tid == 0) {
    float t[5] = { 0.f, 0.f, 0.f, 0.f, 0.f };
#pragma unroll
    for (int w = 0; w < TPB / WAVE; ++w)
#pragma unroll
      for (int k = 0; k < 5; ++k) t[k] += wred[w][k];
    float* out = partials + ((long long)row * BPR + blk) * 5;
#pragma unroll
    for (int k = 0; k < 5; ++k) out[k] = t[k];
  }
}

__global__ void __launch_bounds__(TPB)
pearson_final_kernel(const float* __restrict__ partials,  // [B][BPR][5]
                     float* __restrict__ out,
                     int nrows, float nf) {
  __shared__ float red[TPB];
  const int tid = threadIdx.x;
  float val = 0.f;
  if (tid < nrows) {
    float s[5] = { 0.f, 0.f, 0.f, 0.f, 0.f };
    const float* p = partials + (long long)tid * BPR * 5;
    for (int j = 0; j < BPR; ++j)
#pragma unroll
      for (int k = 0; k < 5; ++k) s[k] += p[j * 5 + k];
    const float num = nf * s[2] - s[0] * s[1];
    const float den = sqrtf((nf * s[3] - s[0] * s[0]) *
                            (nf * s[4] - s[1] * s[1]));
    val = 1.0f - num / den;
  }
  red[tid] = val;
  __syncthreads();
  for (int off = TPB / 2; off > 0; off >>= 1) {
    if (tid < off) red[tid] += red[tid + off];
    __syncthreads();
  }
  if (tid == 0) out[0] = red[0] / (float)nrows;
}

extern "C" void kernel_launch(void* const* d_in, const int* in_sizes, int n_in,
                              void* d_out, int out_size, void* d_ws, size_t ws_size,
                              hipStream_t stream) {
  const float* preds  = (const float*)d_in[0];
  const float* labels = (const float*)d_in[1];
  // d_in[2] = subject scalar, unused by the math.
  float* out = (float*)d_out;

  const int B = 256;                 // fixed by the reference
  const int N = in_sizes[0] / B;     // 100000

  float* partials = (float*)d_ws;    // B*BPR*5 floats = 40 KB, fully overwritten

  dim3 grid(BPR, B);
  pearson_partial_kernel<<<grid, TPB, 0, stream>>>(preds, labels, partials, N);
  pearson_final_kernel<<<1, TPB, 0, stream>>>(partials, out, B, (float)N);
}